// GNNClassifier_4071628996831
// MI455X (gfx1250) — compile-verified
//
#include <hip/hip_runtime.h>
#include <cstddef>

typedef __attribute__((ext_vector_type(2))) float v2f;
typedef __attribute__((ext_vector_type(8))) float v8f;

#define HIDC 128
#define INC  64
#define NGRAPH 64
#define NCLS 2

// ---------------------------------------------------------------------------
// FP32 WMMA GEMM, strip-blocked: each wave computes a 16x128 strip of
// C = (RELU_A ? relu(A) : A)[M,K] @ B[K,128] using 8 accumulators
// (8 x v8f = 64 VGPRs) so the A fragment is loaded once per k-step and
// reused by 8 V_WMMA_F32_16X16X4_F32 ops. M must be a multiple of 16.
// ---------------------------------------------------------------------------
template <bool RELU_A, int K, int NC>
__global__ __launch_bounds__(256) void gemm_wmma_f32_strip(
    const float* __restrict__ A, const float* __restrict__ B,
    float* __restrict__ C, int strips_total) {
  constexpr int NT = NC / 16;  // 8 column tiles
  const int lane = threadIdx.x & 31;
  const int wave = threadIdx.x >> 5;
  const int strip = blockIdx.x * 8 + wave;  // wave-uniform guard
  if (strip >= strips_total) return;

  const int half = lane >> 4;  // 0: lanes 0-15, 1: lanes 16-31
  const int l    = lane & 15;

  // A 16x4 fp32 fragment: M = l, lane-half selects K pair (0/1 vs 2/3).
  const float* __restrict__ arow = A + (size_t)(strip * 16 + l) * K + half * 2;
  // B 4x16 fp32 fragments: N = t*16 + l, symmetric K-pair layout.
  const float* __restrict__ brow = B + (size_t)(half * 2) * NC + l;

  v8f acc[NT] = {};
#pragma unroll 4
  for (int k = 0; k < K; k += 4) {
    v2f a;
    a.x = arow[k + 0];
    a.y = arow[k + 1];
    if (RELU_A) { a.x = fmaxf(a.x, 0.0f); a.y = fmaxf(a.y, 0.0f); }
#pragma unroll
    for (int t = 0; t < NT; ++t) {
      v2f b;
      b.x = brow[(size_t)(k + 0) * NC + t * 16];
      b.y = brow[(size_t)(k + 1) * NC + t * 16];
      // 8 args: (neg_a, A, neg_b, B, c_mod, C, reuse_a, reuse_b)
      acc[t] = __builtin_amdgcn_wmma_f32_16x16x4_f32(
          false, a, false, b, (short)0, acc[t], false, false);
    }
  }

  // C/D layout: VGPR r -> M = r (lanes 0-15) / r+8 (lanes 16-31), N = l.
  float* crow = C + (size_t)(strip * 16 + half * 8) * NC + l;
#pragma unroll
  for (int t = 0; t < NT; ++t)
#pragma unroll
    for (int r = 0; r < 8; ++r) crow[(size_t)r * NC + t * 16] = acc[t][r];
}

// agg[i, ch] = bias[ch] for all nodes; float4-vectorized (32 groups/node).
__global__ void init_bias_kernel(float4* __restrict__ dst,
                                 const float4* __restrict__ bias4,
                                 long long total4 /* N*32 */) {
  long long i = (long long)blockIdx.x * blockDim.x + threadIdx.x;
  if (i < total4) dst[i] = bias4[i & 31];
}

// msg = h[src] * ew; agg[dst] += msg.  One thread per (edge, 4-channel
// group): b128 gathers of h, edge indices/weight loaded once per 4 channels.
// Gathers and atomics land in the 51 MB h/agg working set -> L2 resident.
__global__ void scatter_edges_kernel(const float4* __restrict__ h4,
                                     const int* __restrict__ src,
                                     const int* __restrict__ dstv,
                                     const float* __restrict__ ew,
                                     float* __restrict__ agg,
                                     long long total /* E*32 */) {
  long long t = (long long)blockIdx.x * blockDim.x + threadIdx.x;
  if (t >= total) return;
  const int e  = (int)(t >> 5);
  const int c4 = (int)(t & 31);
  const int s = src[e];
  const int d = dstv[e];
  const float w = ew[e];
  const float4 v = h4[(size_t)s * 32 + c4];
  float* base = agg + (size_t)d * HIDC + c4 * 4;
  atomicAdd(base + 0, v.x * w);
  atomicAdd(base + 1, v.y * w);
  atomicAdd(base + 2, v.z * w);
  atomicAdd(base + 3, v.w * w);
}

__global__ void zero_kernel(float* __restrict__ p, int n) {
  int i = blockIdx.x * blockDim.x + threadIdx.x;
  if (i < n) p[i] = 0.0f;
}

// sums[g, ch] += relu(agg[i, ch]); cnt[g] += 1 (once per node).
__global__ void pool_kernel(const float4* __restrict__ agg4,
                            const int* __restrict__ batch,
                            float* __restrict__ sums, float* __restrict__ cnt,
                            long long total /* N*32 */) {
  long long t = (long long)blockIdx.x * blockDim.x + threadIdx.x;
  if (t >= total) return;
  const int i  = (int)(t >> 5);
  const int c4 = (int)(t & 31);
  const int g = batch[i];
  const float4 v = agg4[t];
  float* base = sums + (size_t)g * HIDC + c4 * 4;
  atomicAdd(base + 0, fmaxf(v.x, 0.0f));
  atomicAdd(base + 1, fmaxf(v.y, 0.0f));
  atomicAdd(base + 2, fmaxf(v.z, 0.0f));
  atomicAdd(base + 3, fmaxf(v.w, 0.0f));
  if (c4 == 0) atomicAdd(&cnt[g], 1.0f);
}

// out[g, c] = (sums[g]/max(cnt,1)) . Wfc[:, c] + bfc[c]  -- 64x2, one block.
__global__ void final_fc_kernel(const float* __restrict__ sums,
                                const float* __restrict__ cnt,
                                const float* __restrict__ Wfc,
                                const float* __restrict__ bfc,
                                float* __restrict__ out) {
  const int t = threadIdx.x;  // 128 threads = 64 graphs x 2 classes
  const int g = t >> 1;
  const int c = t & 1;
  const float inv = 1.0f / fmaxf(cnt[g], 1.0f);
  float acc = bfc[c];
#pragma unroll 8
  for (int ch = 0; ch < HIDC; ++ch)
    acc += sums[(size_t)g * HIDC + ch] * inv * Wfc[ch * NCLS + c];
  out[g * NCLS + c] = acc;
}

extern "C" void kernel_launch(void* const* d_in, const int* in_sizes, int n_in,
                              void* d_out, int out_size, void* d_ws,
                              size_t ws_size, hipStream_t stream) {
  const float* x     = (const float*)d_in[0];
  const int*   ei    = (const int*)d_in[1];    // [2, E]
  const float* ew    = (const float*)d_in[2];  // [E]
  const int*   batch = (const int*)d_in[3];    // [N]
  const float* W1    = (const float*)d_in[4];
  const float* b1    = (const float*)d_in[5];
  const float* W2    = (const float*)d_in[6];
  const float* b2    = (const float*)d_in[7];
  const float* Wfc   = (const float*)d_in[8];
  const float* bfc   = (const float*)d_in[9];
  float* out = (float*)d_out;

  const int N = in_sizes[0] / INC;  // 100000
  const int E = in_sizes[2];        // 1600000
  const int* src = ei;
  const int* dst = ei + E;

  // Workspace: two [N,128] fp32 ping-pong buffers + pooled sums/counts.
  float* buf0 = (float*)d_ws;                    // h_pre / h2_pre
  float* buf1 = buf0 + (size_t)N * HIDC;         // agg1 / agg2
  float* sums = buf1 + (size_t)N * HIDC;         // [64,128]
  float* cnt  = sums + NGRAPH * HIDC;            // [64]

  const long long n4 = (long long)N * 32;        // node 4-channel groups
  const long long e4 = (long long)E * 32;        // edge 4-channel groups
  const int strips = N / 16;                     // 6250
  const int gemm_blocks = (strips + 7) / 8;

  // ---- Layer 1: buf0 = x @ W1 ; buf1 = b1 ; buf1 += scatter(buf0) --------
  gemm_wmma_f32_strip<false, INC, HIDC>
      <<<gemm_blocks, 256, 0, stream>>>(x, W1, buf0, strips);
  init_bias_kernel<<<(int)((n4 + 255) / 256), 256, 0, stream>>>(
      (float4*)buf1, (const float4*)b1, n4);
  scatter_edges_kernel<<<(int)((e4 + 255) / 256), 256, 0, stream>>>(
      (const float4*)buf0, src, dst, ew, buf1, e4);

  // ---- Layer 2: buf0 = relu(buf1) @ W2 ; buf1 = b2 ; buf1 += scatter ----
  gemm_wmma_f32_strip<true, HIDC, HIDC>
      <<<gemm_blocks, 256, 0, stream>>>(buf1, W2, buf0, strips);
  init_bias_kernel<<<(int)((n4 + 255) / 256), 256, 0, stream>>>(
      (float4*)buf1, (const float4*)b2, n4);
  scatter_edges_kernel<<<(int)((e4 + 255) / 256), 256, 0, stream>>>(
      (const float4*)buf0, src, dst, ew, buf1, e4);

  // ---- Mean pool over graphs + final FC ---------------------------------
  const int pz = NGRAPH * HIDC + NGRAPH;
  zero_kernel<<<(pz + 255) / 256, 256, 0, stream>>>(sums, pz);
  pool_kernel<<<(int)((n4 + 255) / 256), 256, 0, stream>>>(
      (const float4*)buf1, batch, sums, cnt, n4);
  final_fc_kernel<<<1, NGRAPH * NCLS, 0, stream>>>(sums, cnt, Wfc, bfc, out);
}